// MatchesLayerDistillationSegmentorSelfV2_84361747628541
// MI455X (gfx1250) — compile-verified
//
#include <hip/hip_runtime.h>
#include <math.h>

typedef float v2f __attribute__((ext_vector_type(2)));
typedef float v8f __attribute__((ext_vector_type(8)));

#define N_PTS 65536
#define M_PTS 65536
#define NCLS  22
#define INV_T 0.5f      // 1/TEMP
#define KL_SCALE 0.8f   // TEMP^2 * KL_W = 4 * 0.2

// ---------------------------------------------------------------------------
// Phase 1: brute-force 1-NN via V_WMMA_F32_16X16X4_F32.
// Each wave owns 16 student rows. K-slot trick: A = [-2sx, -2sy | -2sz, 1.0],
// B = [tx, ty | tz, t2]  =>  D = t2 - 2*dot = d2 - s2 (argmin key), C = inline 0.
// Software-pipelined over pairs of 16-teacher tiles (step 32), unrolled x2 so
// the pipeline's register rotation becomes pure renaming (no v_mov).
// Block = 256 threads = 8 waves = 128 student rows per workgroup.
// ---------------------------------------------------------------------------
__global__ __launch_bounds__(256)
void knn_wmma_kernel(const float* __restrict__ s_coord,
                     const float* __restrict__ t_coord,
                     int* __restrict__ nn_idx)
{
    const int lane = threadIdx.x & 31;
    const int wave = threadIdx.x >> 5;
    const int row0 = blockIdx.x * 128 + wave * 16;   // first of 16 student rows
    const int lr   = lane & 15;
    const bool hi  = (lane >= 16);

    // A-matrix (16x4 f32), constant over the whole teacher loop:
    //   lanes 0-15 : row=lr, (K0,K1) = (-2sx, -2sy)
    //   lanes 16-31: row=lr, (K2,K3) = (-2sz,  1.0)   <- K3 picks up t2 from B
    const float* sp = s_coord + 3 * (row0 + lr);
    const float sx = sp[0], sy = sp[1], sz = sp[2];
    v2f a;
    a.x = hi ? (-2.0f * sz) : (-2.0f * sx);
    a.y = hi ? 1.0f         : (-2.0f * sy);

    float best[8];
    int   bidx[8];
#pragma unroll
    for (int e = 0; e < 8; ++e) { best[e] = 3.4e38f; bidx[e] = 0; }

    // One 16-teacher tile: build B, WMMA with inline-0 C, compare/select.
    auto process = [&](float tx, float ty, float tz, int candv) {
        const float t2 = tx * tx + ty * ty + tz * tz;
        v2f b;
        b.x = hi ? tz : tx;
        b.y = hi ? t2 : ty;       // K3 row of B carries t2
        v8f c = {};               // folds to inline SRC2 = 0
        v8f d = __builtin_amdgcn_wmma_f32_16x16x4_f32(
            false, a, false, b, (short)0, c, false, false);
#pragma unroll
        for (int e = 0; e < 8; ++e) {
            if (d[e] < best[e]) { best[e] = d[e]; bidx[e] = candv; }
        }
    };

    // ---- software pipeline: preload tile pair, then load-ahead each step ----
    const float* p0 = t_coord + 3 * lr;
    const float* p1 = t_coord + 3 * (16 + lr);
    float cx0 = p0[0], cy0 = p0[1], cz0 = p0[2];
    float cx1 = p1[0], cy1 = p1[1], cz1 = p1[2];

    int cand = lr;
#pragma unroll 2
    for (int j0 = 0; j0 < M_PTS - 32; j0 += 32) {
        const float* n0 = t_coord + 3 * (j0 + 32 + lr);
        const float* n1 = t_coord + 3 * (j0 + 48 + lr);
        const float nx0 = n0[0], ny0 = n0[1], nz0 = n0[2];
        const float nx1 = n1[0], ny1 = n1[1], nz1 = n1[2];
        __builtin_prefetch(n0 + 3 * 512, 0, 1);   // stream teachers ahead

        process(cx0, cy0, cz0, cand);
        process(cx1, cy1, cz1, cand + 16);
        cand += 32;

        cx0 = nx0; cy0 = ny0; cz0 = nz0;
        cx1 = nx1; cy1 = ny1; cz1 = nz1;
    }
    process(cx0, cy0, cz0, cand);
    process(cx1, cy1, cz1, cand + 16);

    // Reduce the 16 teacher columns per student row: branchless butterfly
    // inside each 16-lane half (xor masks 8..1 never cross halves on wave32).
    // Tie-break on smaller teacher index to match argmin semantics.
#pragma unroll
    for (int e = 0; e < 8; ++e) {
#pragma unroll
        for (int m = 8; m >= 1; m >>= 1) {
            float ov = __shfl_xor(best[e], m, 32);
            int   oi = __shfl_xor(bidx[e], m, 32);
            int better = (int)(ov < best[e]) |
                         ((int)(ov == best[e]) & (int)(oi < bidx[e]));
            best[e] = better ? ov : best[e];
            bidx[e] = better ? oi : bidx[e];
        }
    }

    if (lr == 0) {
        const int rbase = row0 + (hi ? 8 : 0);   // lane0 -> rows 0..7, lane16 -> rows 8..15
#pragma unroll
        for (int e = 0; e < 8; ++e) nn_idx[rbase + e] = bidx[e];
    }
}

// ---------------------------------------------------------------------------
// Phase 2: masked KL(T=2) with batchmean, one thread per student row.
// ---------------------------------------------------------------------------
__global__ __launch_bounds__(256)
void kl_kernel(const float* __restrict__ s_coord,
               const float* __restrict__ t_coord,
               const float* __restrict__ s_logits,
               const float* __restrict__ t_logits,
               const int* __restrict__ nn_idx,
               float* __restrict__ acc)
{
    const int i = blockIdx.x * blockDim.x + threadIdx.x;
    float per = 0.0f, msk = 0.0f;
    if (i < N_PTS) {
        const int col = nn_idx[i];
        const float dx = s_coord[3 * i + 0] - t_coord[3 * col + 0];
        const float dy = s_coord[3 * i + 1] - t_coord[3 * col + 1];
        const float dz = s_coord[3 * i + 2] - t_coord[3 * col + 2];
        const float d2 = dx * dx + dy * dy + dz * dz;
        msk = (sqrtf(d2) <= 0.05f) ? 1.0f : 0.0f;

        const float* sl = s_logits + (size_t)i * NCLS;
        const float* tl = t_logits + (size_t)col * NCLS;

        float sv[NCLS], tv[NCLS];
        float smax = -3.4e38f, tmax = -3.4e38f;
#pragma unroll
        for (int c = 0; c < NCLS; ++c) {
            sv[c] = sl[c] * INV_T;
            tv[c] = tl[c] * INV_T;
            smax = fmaxf(smax, sv[c]);
            tmax = fmaxf(tmax, tv[c]);
        }
        float ssum = 0.0f, tsum = 0.0f;
#pragma unroll
        for (int c = 0; c < NCLS; ++c) {
            ssum += expf(sv[c] - smax);
            tsum += expf(tv[c] - tmax);
        }
        const float slz = smax + logf(ssum);
        const float tlz = tmax + logf(tsum);

        float s = 0.0f;
#pragma unroll
        for (int c = 0; c < NCLS; ++c) {
            const float tlp = tv[c] - tlz;
            const float tpv = expf(tlp);
            s += tpv * (tlp - (sv[c] - slz));
        }
        per = s * msk;
    }

    // wave32 reduction, then one atomic per wave
#pragma unroll
    for (int m = 16; m >= 1; m >>= 1) {
        per += __shfl_xor(per, m, 32);
        msk += __shfl_xor(msk, m, 32);
    }
    if ((threadIdx.x & 31) == 0) {
        atomicAdd(&acc[0], per);
        atomicAdd(&acc[1], msk);
    }
}

__global__ void init_acc_kernel(float* acc)
{
    if (threadIdx.x < 2) acc[threadIdx.x] = 0.0f;
}

__global__ void finish_kernel(const float* __restrict__ acc, float* __restrict__ out)
{
    const float n  = acc[1];
    const float kl = (n > 0.0f) ? acc[0] / fmaxf(n, 1.0f) : 0.0f;
    out[0] = kl * KL_SCALE;
}

// ---------------------------------------------------------------------------
extern "C" void kernel_launch(void* const* d_in, const int* in_sizes, int n_in,
                              void* d_out, int out_size, void* d_ws, size_t ws_size,
                              hipStream_t stream)
{
    const float* s_coord  = (const float*)d_in[0];
    const float* t_coord  = (const float*)d_in[1];
    const float* s_logits = (const float*)d_in[2];
    const float* t_logits = (const float*)d_in[3];
    float* out = (float*)d_out;

    int*   nn  = (int*)d_ws;                                          // N ints
    float* acc = (float*)((char*)d_ws + (size_t)N_PTS * sizeof(int)); // 2 floats

    init_acc_kernel<<<1, 32, 0, stream>>>(acc);
    knn_wmma_kernel<<<N_PTS / 128, 256, 0, stream>>>(s_coord, t_coord, nn);
    kl_kernel<<<N_PTS / 256, 256, 0, stream>>>(s_coord, t_coord, s_logits, t_logits, nn, acc);
    finish_kernel<<<1, 1, 0, stream>>>(acc, out);
}